// LightGCN_52862457479751
// MI455X (gfx1250) — compile-verified
//
#include <hip/hip_runtime.h>
#include <hip/hip_bf16.h>
#include <stdint.h>
#include <math.h>

// ---------------------------------------------------------------------------
// LightGCN on MI455X (gfx1250, wave32).
// CSR-by-destination build (no float atomics), then 3 pull-style layers:
// wave32 per node, float2 per lane, double-buffered async-to-LDS edge staging
// (ASYNCcnt pipelining: stage chunk c+1, s_wait_asynccnt 2, process chunk c).
// ---------------------------------------------------------------------------

#define EMB 64
#define SCAN_BLK 256
#define WAVES_PER_BLK 8
#define CHUNK 32

#if defined(__has_builtin)
#if __has_builtin(__builtin_amdgcn_global_load_async_to_lds_b32)
#define HAVE_ASYNC_LDS 1
#endif
#endif
#ifndef HAVE_ASYNC_LDS
#define HAVE_ASYNC_LDS 0
#endif

typedef __attribute__((address_space(1))) int gint_t;   // global (AS1) int
typedef __attribute__((address_space(3))) int lint_t;   // LDS (AS3) int

template <int N>
static __device__ __forceinline__ void wait_async() {
#if defined(__has_builtin) && __has_builtin(__builtin_amdgcn_s_wait_asynccnt)
    __builtin_amdgcn_s_wait_asynccnt(N);
#else
    asm volatile("s_wait_asynccnt %0" ::"i"(N) : "memory");
#endif
}

// ---- 1. zero the degree/count array -----------------------------------------
__global__ void k_zero_i32(int* __restrict__ p, int n) {
    int i = blockIdx.x * blockDim.x + threadIdx.x;
    if (i < n) p[i] = 0;
}

// ---- 2. degree count over col ----------------------------------------------
__global__ void k_count(const int* __restrict__ col, int* __restrict__ counts, int nE) {
    int e = blockIdx.x * blockDim.x + threadIdx.x;
    if (e < nE) atomicAdd(&counts[col[e]], 1);
}

// ---- 3a. per-block exclusive scan ------------------------------------------
__global__ __launch_bounds__(SCAN_BLK) void k_scan_block(const int* __restrict__ counts,
                                                         int* __restrict__ rowptr,
                                                         int* __restrict__ bsum, int n) {
    __shared__ int s[SCAN_BLK];
    int tid = threadIdx.x;
    int i = blockIdx.x * SCAN_BLK + tid;
    int v = (i < n) ? counts[i] : 0;
    s[tid] = v;
    __syncthreads();
    for (int off = 1; off < SCAN_BLK; off <<= 1) {
        int t = (tid >= off) ? s[tid - off] : 0;
        __syncthreads();
        if (tid >= off) s[tid] += t;
        __syncthreads();
    }
    if (i < n) rowptr[i] = s[tid] - v;          // exclusive
    if (tid == SCAN_BLK - 1) bsum[blockIdx.x] = s[tid];
}

// ---- 3b. scan of block sums (nb <= 1024) -----------------------------------
__global__ __launch_bounds__(1024) void k_scan_top(int* __restrict__ bsum, int nb) {
    __shared__ int s[1024];
    int tid = threadIdx.x;
    int v = (tid < nb) ? bsum[tid] : 0;
    s[tid] = v;
    __syncthreads();
    for (int off = 1; off < 1024; off <<= 1) {
        int t = (tid >= off) ? s[tid - off] : 0;
        __syncthreads();
        if (tid >= off) s[tid] += t;
        __syncthreads();
    }
    if (tid < nb) bsum[tid] = s[tid] - v;       // exclusive
}

// ---- 3c. finalize rowptr, cursor, deg^-1/2 ---------------------------------
__global__ void k_finalize(int* __restrict__ rowptr, const int* __restrict__ bsum,
                           int* __restrict__ cursor, const int* __restrict__ counts,
                           float* __restrict__ dis, int n) {
    int i = blockIdx.x * blockDim.x + threadIdx.x;
    if (i >= n) return;
    int rp = rowptr[i] + bsum[i >> 8];
    rowptr[i] = rp;
    cursor[i] = rp;
    float d = (float)counts[i];
    dis[i] = (d > 0.0f) ? (1.0f / sqrtf(d)) : 0.0f;
}

// ---- 4. CSR build: place (src,norm) grouped by destination -----------------
__global__ void k_build(const int* __restrict__ row, const int* __restrict__ col,
                        int* __restrict__ cursor, const float* __restrict__ dis,
                        int* __restrict__ csrc, float* __restrict__ cnrm, int nE) {
    int e = blockIdx.x * blockDim.x + threadIdx.x;
    if (e >= nE) return;
    int r = row[e];
    int c = col[e];
    int pos = atomicAdd(&cursor[c], 1);
    csrc[pos] = r;
    cnrm[pos] = dis[r] * dis[c];
}

// ---- 5. init embA and the layer accumulator (d_out) ------------------------
__global__ void k_init(const float* __restrict__ w, float* __restrict__ embA,
                       float* __restrict__ out, int n) {
    int i = blockIdx.x * blockDim.x + threadIdx.x;
    if (i < n) {
        float v = w[i];
        embA[i] = v;
        out[i] = v;
    }
}

// ---- 6. propagation layer: wave32 per destination node ---------------------
__global__ __launch_bounds__(WAVES_PER_BLK * 32) void k_layer(
    const float* __restrict__ embIn, float* __restrict__ embOut,
    float* __restrict__ outAcc,
    const int* __restrict__ rowptr, const int* __restrict__ counts,
    const int* __restrict__ csrc, const float* __restrict__ cnrm, int nNodes) {
    // interleaved (src, norm-bits) pairs, double-buffered per wave
    __shared__ int2 s_pair[WAVES_PER_BLK][2][CHUNK];

    const int lane = threadIdx.x & 31;
    const int wid = threadIdx.x >> 5;
    const int node = blockIdx.x * WAVES_PER_BLK + wid;
    if (node >= nNodes) return;

    const int beg = __builtin_amdgcn_readfirstlane(rowptr[node]);
    const int cnt = __builtin_amdgcn_readfirstlane(counts[node]);

    const float2* __restrict__ in2 = (const float2*)embIn;
    float accx = 0.0f, accy = 0.0f;

    // stage one CHUNK of the edge list into LDS buffer `buf`
    auto stage = [&](int cbase, int buf) {
        const int idx = beg + cbase + lane;
        const int rem = cnt - cbase;
        if (lane == 0 && cbase + CHUNK < cnt) {           // global_prefetch_b8
            __builtin_prefetch(csrc + beg + cbase + CHUNK, 0, 0);
            __builtin_prefetch(cnrm + beg + cbase + CHUNK, 0, 0);
        }
        if (lane < rem) {
#if HAVE_ASYNC_LDS
            __builtin_amdgcn_global_load_async_to_lds_b32(
                (gint_t*)(csrc + idx), (lint_t*)&s_pair[wid][buf][lane].x, 0, 0);
            __builtin_amdgcn_global_load_async_to_lds_b32(
                (gint_t*)(cnrm + idx), (lint_t*)&s_pair[wid][buf][lane].y, 0, 0);
#else
            s_pair[wid][buf][lane].x = csrc[idx];
            s_pair[wid][buf][lane].y = __float_as_int(cnrm[idx]);
#endif
        }
    };

    const int nChunks = (cnt + CHUNK - 1) >> 5;
    if (nChunks > 0) stage(0, 0);

    for (int c = 0; c < nChunks; ++c) {
        const int buf = c & 1;
        const int cbase = c * CHUNK;
        const int rem = (cnt - cbase < CHUNK) ? (cnt - cbase) : CHUNK;

        if (c + 1 < nChunks) {
            stage(cbase + CHUNK, buf ^ 1);   // next chunk in flight...
            wait_async<2>();                 // ...while waiting only on chunk c
        } else {
            wait_async<0>();
        }
        __builtin_amdgcn_wave_barrier();     // wave-synchronous LDS exchange

#pragma unroll 4
        for (int i = 0; i < rem; ++i) {
            const int2 pr = s_pair[wid][buf][i];           // ds_load_b64
            const int src = pr.x;
            const float wgt = __int_as_float(pr.y);
            const float2 v = in2[(size_t)src * (EMB / 2) + lane];
            accx = fmaf(wgt, v.x, accx);
            accy = fmaf(wgt, v.y, accy);
        }
        __builtin_amdgcn_wave_barrier();
    }

    const size_t o = (size_t)node * (EMB / 2) + lane;
    float2* out2 = (float2*)embOut;
    float2 r;
    r.x = accx;
    r.y = accy;
    out2[o] = r;

    float2* acc2 = (float2*)outAcc;
    float2 p = acc2[o];
    p.x += accx;
    p.y += accy;
    acc2[o] = p;
}

// ---- 7. final mean over (layers+1) -----------------------------------------
__global__ void k_scale(float* __restrict__ out, float s, int n) {
    int i = blockIdx.x * blockDim.x + threadIdx.x;
    if (i < n) out[i] *= s;
}

// ---------------------------------------------------------------------------
extern "C" void kernel_launch(void* const* d_in, const int* in_sizes, int n_in,
                              void* d_out, int out_size, void* d_ws, size_t ws_size,
                              hipStream_t stream) {
    const int* edge = (const int*)d_in[0];
    const float* embw = (const float*)d_in[1];
    const int nE = in_sizes[0] / 2;
    const int nN = in_sizes[1] / EMB;
    const int* row = edge;
    const int* col = edge + nE;
    float* out = (float*)d_out;

    // workspace carve-out (256B aligned)
    char* ws = (char*)d_ws;
    size_t off = 0;
    auto take = [&](size_t bytes) -> char* {
        char* p = ws + off;
        off = (off + bytes + 255) & ~(size_t)255;
        return p;
    };
    int* counts = (int*)take((size_t)nN * 4);
    int* rowptr = (int*)take((size_t)nN * 4);
    int* cursor = (int*)take((size_t)nN * 4);
    float* dis = (float*)take((size_t)nN * 4);
    int* bsum = (int*)take(1024 * 4);
    int* csrc = (int*)take((size_t)nE * 4);
    float* cnrm = (float*)take((size_t)nE * 4);
    float* embA = (float*)take((size_t)nN * EMB * 4);
    float* embB = (float*)take((size_t)nN * EMB * 4);

    const int nb = (nN + SCAN_BLK - 1) / SCAN_BLK;   // 586 for N=150000 (<1024)
    const int gN = (nN + 255) / 256;
    const int gE = (nE + 255) / 256;
    const int gV = (nN * EMB + 255) / 256;
    const int gL = (nN + WAVES_PER_BLK - 1) / WAVES_PER_BLK;

    // degree + CSR build
    k_zero_i32<<<gN, 256, 0, stream>>>(counts, nN);
    k_count<<<gE, 256, 0, stream>>>(col, counts, nE);
    k_scan_block<<<nb, SCAN_BLK, 0, stream>>>(counts, rowptr, bsum, nN);
    k_scan_top<<<1, 1024, 0, stream>>>(bsum, nb);
    k_finalize<<<gN, 256, 0, stream>>>(rowptr, bsum, cursor, counts, dis, nN);
    k_build<<<gE, 256, 0, stream>>>(row, col, cursor, dis, csrc, cnrm, nE);

    // layer 0 state
    k_init<<<gV, 256, 0, stream>>>(embw, embA, out, nN * EMB);

    // 3 propagation layers (ping-pong)
    k_layer<<<gL, WAVES_PER_BLK * 32, 0, stream>>>(embA, embB, out, rowptr, counts, csrc, cnrm, nN);
    k_layer<<<gL, WAVES_PER_BLK * 32, 0, stream>>>(embB, embA, out, rowptr, counts, csrc, cnrm, nN);
    k_layer<<<gL, WAVES_PER_BLK * 32, 0, stream>>>(embA, embB, out, rowptr, counts, csrc, cnrm, nN);

    // mean over 4 terms
    k_scale<<<gV, 256, 0, stream>>>(out, 0.25f, nN * EMB);
}